// GIN_31104153158276
// MI455X (gfx1250) — compile-verified
//
#include <hip/hip_runtime.h>
#include <hip/hip_bf16.h>
#include <math.h>

typedef __attribute__((ext_vector_type(2))) float v2f;
typedef __attribute__((ext_vector_type(8))) float v8f;

#define GIN_N   50000
#define GIN_E   800000
#define GIN_FIN 100
#define GIN_H   64
#define GIN_C   18

// ---------------------------------------------------------------------------
// CDNA5 async global->LDS copy (16B per lane), tracked by ASYNCcnt.
// Inline asm per bridge doc (portable across toolchains).
// ---------------------------------------------------------------------------
__device__ __forceinline__ unsigned lds_off(const void* p) {
  // flat LDS address: addr[31:0] == workgroup-relative LDS byte offset
  return (unsigned)(uintptr_t)p;
}
__device__ __forceinline__ void async_g2l_b128(unsigned ldsOff, const void* g) {
  asm volatile("global_load_async_to_lds_b128 %0, %1, off"
               :: "v"(ldsOff), "v"(g) : "memory");
}
__device__ __forceinline__ void wait_async0() {
  asm volatile("s_wait_asynccnt 0" ::: "memory");
}

// ---------------------------------------------------------------------------
// Z = (1+eps) * X   (replaces memset of the aggregation buffer; scatter-add
// then accumulates straight into Z, so the MLP tile fill is a pure copy).
// ---------------------------------------------------------------------------
__global__ __launch_bounds__(256) void scale_init(
    const float* __restrict__ X, float* __restrict__ Z,
    const float* __restrict__ epsArr, int epsIdx, long count4)
{
  const float s = 1.0f + epsArr[epsIdx];
  const float4* xi = (const float4*)X;
  float4* zo = (float4*)Z;
  long i = (long)blockIdx.x * blockDim.x + threadIdx.x;
  const long stride = (long)gridDim.x * blockDim.x;
  for (; i < count4; i += stride) {
    float4 v = xi[i];
    v.x *= s; v.y *= s; v.z *= s; v.w *= s;
    zo[i] = v;
  }
}

// ---------------------------------------------------------------------------
// Edge-parallel scatter-add: Z[dst] += X[src].  One wave per edge, lanes
// sweep the feature dim.  All operands are L2-resident (192MB L2).
// ---------------------------------------------------------------------------
__global__ __launch_bounds__(256) void scatter_add_f(
    const float* __restrict__ X, const int* __restrict__ src,
    const int* __restrict__ dst, float* __restrict__ Z,
    int nEdges, int F)
{
  const int lane   = threadIdx.x & 31;
  const int wave   = (blockIdx.x << 3) + (threadIdx.x >> 5);
  const int stride = gridDim.x << 3;
  for (int e = wave; e < nEdges; e += stride) {
    const int s = src[e];
    const int d = dst[e];
    const float* xr = X + (size_t)s * F;
    float* zr       = Z + (size_t)d * F;
    for (int f = lane; f < F; f += 32)
      atomicAdd(zr + f, xr[f]);
  }
}

// ---------------------------------------------------------------------------
// Wave-level 16xK * Kx(NT*16) fp32 GEMM using V_WMMA_F32_16X16X4_F32.
// ---------------------------------------------------------------------------
template<int K, int AS, int WS, int NT>
__device__ __forceinline__ void wave_gemm(const float* __restrict__ sA,
                                          const float* __restrict__ sW,
                                          v8f* acc)
{
  const int lane = threadIdx.x & 31;
  const int m    = lane & 15;
  const int half = lane >> 4;
  for (int k = 0; k < K; k += 4) {
    v2f a;
    a.x = sA[m * AS + k + 2 * half];
    a.y = sA[m * AS + k + 2 * half + 1];
#pragma unroll
    for (int t = 0; t < NT; ++t) {
      v2f b;
      b.x = sW[(k + 2 * half)     * WS + t * 16 + m];
      b.y = sW[(k + 2 * half + 1) * WS + t * 16 + m];
      acc[t] = __builtin_amdgcn_wmma_f32_16x16x4_f32(
          false, a, false, b, (short)0, acc[t], false, false);
    }
  }
}

// ---------------------------------------------------------------------------
// Fused GINConv MLP block, 64 nodes / workgroup (4 waves x 16 rows):
//   Z tile + W1 + W2 staged via global_load_async_to_lds_b128
//   t = ReLU(Z @ W1 + b1)  (WMMA, written back into the Z tile)
//   h = BN(ReLU(t @ W2 + b2))
// ---------------------------------------------------------------------------
template<int K, int ASTR>
__global__ __launch_bounds__(128) void gin_mlp_bn(
    const float* __restrict__ Z,
    const float* __restrict__ W1, const float* __restrict__ B1,
    const float* __restrict__ W2, const float* __restrict__ B2,
    const float* __restrict__ bnG, const float* __restrict__ bnB,
    const float* __restrict__ bnM, const float* __restrict__ bnV,
    float* __restrict__ OUT, int n)
{
  constexpr int WS  = 80;   // 320B rows: 16B aligned, halves hit disjoint banks
  constexpr int TS  = 68;   // padded T stride (272B, 16B aligned)
  constexpr int K4  = K / 4;
  constexpr int ASZ = (64 * ASTR > 64 * TS) ? 64 * ASTR : 64 * TS;

  __shared__ __align__(16) float sA[ASZ];       // Z tile, reused for T
  __shared__ __align__(16) float sW1[K * WS];
  __shared__ __align__(16) float sW2[64 * WS];
  __shared__ float sB1[64];
  __shared__ float sB2[64];
  __shared__ float sScale[64];
  __shared__ float sShift[64];

  const int tid  = threadIdx.x;
  const int row0 = blockIdx.x * 64;

  // async-stage weight tiles (pure copies, 16B granules)
  for (int i = tid; i < K * 16; i += 128) {
    const int r = i >> 4, c4 = (i & 15) * 4;
    async_g2l_b128(lds_off(&sW1[r * WS + c4]), W1 + r * 64 + c4);
  }
  for (int i = tid; i < 64 * 16; i += 128) {
    const int r = i >> 4, c4 = (i & 15) * 4;
    async_g2l_b128(lds_off(&sW2[r * WS + c4]), W2 + r * 64 + c4);
  }
  // async-stage Z tile (already holds (1+eps)x + agg)
  for (int i = tid; i < 64 * K4; i += 128) {
    const int r = i / K4, c4 = (i % K4) * 4;
    int gr = row0 + r; if (gr >= n) gr = n - 1;
    async_g2l_b128(lds_off(&sA[r * ASTR + c4]), Z + (size_t)gr * K + c4);
  }
  if (tid < 64) {
    sB1[tid] = B1[tid];
    sB2[tid] = B2[tid];
    const float sc = bnG[tid] * rsqrtf(bnV[tid] + 1e-5f);
    sScale[tid] = sc;
    sShift[tid] = bnB[tid] - bnM[tid] * sc;
  }
  // hint the next block's Z tile toward L2 while we compute
  __builtin_prefetch(Z + (size_t)(row0 + 64) * K + tid * 16, 0, 0);
  wait_async0();
  __syncthreads();

  const int w    = tid >> 5;
  const int lane = tid & 31;
  const int m    = lane & 15;
  const int half = lane >> 4;

  // ---- stage 1: T = ReLU(Z @ W1 + b1) ----
  v8f acc[4] = {};
  wave_gemm<K, ASTR, WS, 4>(sA + (w * 16) * ASTR, sW1, acc);
  __syncthreads();   // all waves done reading Z before re-striding sA as T
#pragma unroll
  for (int t = 0; t < 4; ++t) {
    const int col = t * 16 + m;
    const float b = sB1[col];
#pragma unroll
    for (int j = 0; j < 8; ++j) {
      const int r = w * 16 + j + 8 * half;
      float v = acc[t][j] + b;
      sA[r * TS + col] = v > 0.0f ? v : 0.0f;
    }
  }

  // ---- stage 2: H = BN(ReLU(T @ W2 + b2)) ----
  v8f acc2[4] = {};
  wave_gemm<64, TS, WS, 4>(sA + (w * 16) * TS, sW2, acc2);
#pragma unroll
  for (int t = 0; t < 4; ++t) {
    const int col = t * 16 + m;
    const float b2v = sB2[col], sc = sScale[col], sh = sShift[col];
#pragma unroll
    for (int j = 0; j < 8; ++j) {
      const int gr = row0 + w * 16 + j + 8 * half;
      if (gr < n) {
        float v = acc2[t][j] + b2v;
        v = v > 0.0f ? v : 0.0f;
        OUT[(size_t)gr * 64 + col] = v * sc + sh;
      }
    }
  }
}

// ---------------------------------------------------------------------------
// Readout: t = ReLU(h @ lin1 + b1); o = t @ lin2 + b2; row log_softmax.
// ---------------------------------------------------------------------------
__global__ __launch_bounds__(128) void readout_kernel(
    const float* __restrict__ Hin,
    const float* __restrict__ W1, const float* __restrict__ B1,
    const float* __restrict__ W2, const float* __restrict__ B2,
    float* __restrict__ OUT, int n)
{
  constexpr int WS = 80, TS = 68, LS = 33;
  __shared__ __align__(16) float sA[64 * TS];
  __shared__ __align__(16) float sW1[64 * WS];
  __shared__ __align__(16) float sW2[64 * WS];
  __shared__ float sB1[64];
  __shared__ float sB2[32];
  __shared__ float sL[64 * LS];

  const int tid  = threadIdx.x;
  const int row0 = blockIdx.x * 64;

  for (int i = tid; i < 64 * 16; i += 128) {
    const int r = i >> 4, c4 = (i & 15) * 4;
    async_g2l_b128(lds_off(&sW1[r * WS + c4]), W1 + r * 64 + c4);
  }
  for (int i = tid; i < 64 * 16; i += 128) {
    const int r = i >> 4, c4 = (i & 15) * 4;
    int gr = row0 + r; if (gr >= n) gr = n - 1;
    async_g2l_b128(lds_off(&sA[r * TS + c4]), Hin + (size_t)gr * 64 + c4);
  }
  // lin2 is 64x18 (rows not 16B): plain fill, zero-padded to 32 cols
  for (int i = tid; i < 64 * 32; i += 128) {
    const int r = i >> 5, c = i & 31;
    sW2[r * WS + c] = (c < GIN_C) ? W2[r * GIN_C + c] : 0.0f;
  }
  if (tid < 64) sB1[tid] = B1[tid];
  if (tid < 32) sB2[tid] = (tid < GIN_C) ? B2[tid] : 0.0f;
  wait_async0();
  __syncthreads();

  const int w = tid >> 5, lane = tid & 31, m = lane & 15, half = lane >> 4;

  v8f acc[4] = {};
  wave_gemm<64, TS, WS, 4>(sA + (w * 16) * TS, sW1, acc);
#pragma unroll
  for (int t = 0; t < 4; ++t) {
    const int col = t * 16 + m;
    const float b = sB1[col];
#pragma unroll
    for (int j = 0; j < 8; ++j) {
      const int r = w * 16 + j + 8 * half;
      float v = acc[t][j] + b;
      sA[r * TS + col] = v > 0.0f ? v : 0.0f;
    }
  }

  v8f acc2[2] = {};
  wave_gemm<64, TS, WS, 2>(sA + (w * 16) * TS, sW2, acc2);
#pragma unroll
  for (int t = 0; t < 2; ++t) {
    const int col = t * 16 + m;
    const float b = sB2[col];
#pragma unroll
    for (int j = 0; j < 8; ++j) {
      const int r = w * 16 + j + 8 * half;
      sL[r * LS + col] = acc2[t][j] + b;
    }
  }
  __syncthreads();

  if (tid < 64) {
    const int gr = row0 + tid;
    if (gr < n) {
      const float* Lr = sL + tid * LS;
      float mx = Lr[0];
      for (int c = 1; c < GIN_C; ++c) mx = fmaxf(mx, Lr[c]);
      float s = 0.0f;
      for (int c = 0; c < GIN_C; ++c) s += expf(Lr[c] - mx);
      const float lse = mx + logf(s);
      for (int c = 0; c < GIN_C; ++c)
        OUT[(size_t)gr * GIN_C + c] = Lr[c] - lse;
    }
  }
}

// ---------------------------------------------------------------------------
extern "C" void kernel_launch(void* const* d_in, const int* in_sizes, int n_in,
                              void* d_out, int out_size, void* d_ws, size_t ws_size,
                              hipStream_t stream)
{
  (void)in_sizes; (void)n_in; (void)out_size; (void)ws_size;

  const float* x   = (const float*)d_in[0];
  const int*   ei  = (const int*)d_in[1];
  const float* eps = (const float*)d_in[2];
  const float* fW1 = (const float*)d_in[3];
  const float* fb1 = (const float*)d_in[4];
  const float* fW2 = (const float*)d_in[5];
  const float* fb2 = (const float*)d_in[6];
  const float* W1s = (const float*)d_in[7];
  const float* b1s = (const float*)d_in[8];
  const float* W2s = (const float*)d_in[9];
  const float* b2s = (const float*)d_in[10];
  const float* bnG = (const float*)d_in[11];
  const float* bnB = (const float*)d_in[12];
  const float* bnM = (const float*)d_in[13];
  const float* bnV = (const float*)d_in[14];
  const float* l1W = (const float*)d_in[15];
  const float* l1b = (const float*)d_in[16];
  const float* l2W = (const float*)d_in[17];
  const float* l2b = (const float*)d_in[18];

  const int N = GIN_N, E = GIN_E;
  const int* src = ei;
  const int* dst = ei + E;

  float* z  = (float*)d_ws;                       // N*100 floats (reused @64)
  float* hA = z + (size_t)N * GIN_FIN;
  float* hB = hA + (size_t)N * GIN_H;

  const int blocksMLP = (N + 63) / 64;
  const int blocksSc  = 2048;

  // first_conv (K = 100)
  scale_init<<<2048, 256, 0, stream>>>(x, z, eps, 0, (long)N * GIN_FIN / 4);
  scatter_add_f<<<blocksSc, 256, 0, stream>>>(x, src, dst, z, E, GIN_FIN);
  gin_mlp_bn<GIN_FIN, GIN_FIN><<<blocksMLP, 128, 0, stream>>>(
      z, fW1, fb1, fW2, fb2,
      bnG + 0, bnB + 0, bnM + 0, bnV + 0, hA, N);

  // layer 0 (K = 64)
  scale_init<<<2048, 256, 0, stream>>>(hA, z, eps, 1, (long)N * GIN_H / 4);
  scatter_add_f<<<blocksSc, 256, 0, stream>>>(hA, src, dst, z, E, GIN_H);
  gin_mlp_bn<GIN_H, 68><<<blocksMLP, 128, 0, stream>>>(
      z, W1s, b1s, W2s, b2s,
      bnG + 64, bnB + 64, bnM + 64, bnV + 64, hB, N);

  // layer 1 (K = 64)
  scale_init<<<2048, 256, 0, stream>>>(hB, z, eps, 2, (long)N * GIN_H / 4);
  scatter_add_f<<<blocksSc, 256, 0, stream>>>(hB, src, dst, z, E, GIN_H);
  gin_mlp_bn<GIN_H, 68><<<blocksMLP, 128, 0, stream>>>(
      z, W1s + GIN_H * GIN_H, b1s + GIN_H,
      W2s + GIN_H * GIN_H, b2s + GIN_H,
      bnG + 128, bnB + 128, bnM + 128, bnV + 128, hA, N);

  // readout + log_softmax
  readout_kernel<<<blocksMLP, 128, 0, stream>>>(
      hA, l1W, l1b, l2W, l2b, (float*)d_out, N);
}